// Transducer_joint_74981539053948
// MI455X (gfx1250) — compile-verified
//
#include <hip/hip_runtime.h>

typedef __attribute__((ext_vector_type(2))) float v2f;
typedef __attribute__((ext_vector_type(4))) float v4f;
typedef __attribute__((ext_vector_type(8))) float v8f;

#define HD 512
#define OD 640
#define TD 300
#define UD 80
#define BD 8

// C[M x 640] = A[M x 512] * W[512 x 640] (+ bias), f32 WMMA 16x16x4.
// 128 threads = 4 waves; block tile 16(M) x 64(N); K staged in 32-wide LDS tiles.
__global__ __launch_bounds__(128)
void gemm_wmma_f32(const float* __restrict__ A,
                   const float* __restrict__ W,
                   const float* __restrict__ bias,   // may be null
                   float* __restrict__ C)
{
    __shared__ float As[16][36];   // 16x32 tile, stride 36 (16B aligned, conflict-free)
    __shared__ float Bs[32][80];   // 32x64 tile, stride 80 (16B aligned, lane-half split)

    const int tid  = threadIdx.x;
    const int lane = tid & 31;
    const int wave = tid >> 5;            // 0..3 -> N sub-tile
    const int m0    = blockIdx.y * 16;
    const int nBase = blockIdx.x * 64;

    const int mloc = lane & 15;           // M (A) / N (B) position
    const int hi   = lane >> 4;           // lane-half -> K offset 0/2

    v8f acc = {};

    for (int k0 = 0; k0 < HD; k0 += 32) {
        // A tile 16x32: one v4f per thread, 128B contiguous per row-group
        {
            const int r  = tid >> 3;
            const int c4 = (tid & 7) << 2;
            *(v4f*)(&As[r][c4]) = *(const v4f*)(A + (m0 + r) * HD + k0 + c4);
        }
        // B tile 32x64: four v4f per thread, 256B contiguous runs
        #pragma unroll
        for (int j = 0; j < 4; ++j) {
            const int f  = tid + 128 * j;   // 0..511
            const int r  = f >> 4;          // 0..31
            const int c4 = (f & 15) << 2;   // 0..60
            *(v4f*)(&Bs[r][c4]) = *(const v4f*)(W + (k0 + r) * OD + nBase + c4);
        }
        __syncthreads();

        #pragma unroll
        for (int kk = 0; kk < 32; kk += 4) {
            v2f a, b;
            // A 16x4 frag: VGPR v = A[m][kk + 2*hi + v]  (contiguous -> ds b64)
            a.x = As[mloc][kk + 2 * hi + 0];
            a.y = As[mloc][kk + 2 * hi + 1];
            // B 4x16 frag: VGPR v = W[kk + 2*hi + v][n]
            b.x = Bs[kk + 2 * hi + 0][wave * 16 + mloc];
            b.y = Bs[kk + 2 * hi + 1][wave * 16 + mloc];
            acc = __builtin_amdgcn_wmma_f32_16x16x4_f32(
                false, a, false, b, (short)0, acc, false, false);
        }
        __syncthreads();
    }

    // C/D layout: VGPR r -> row (r + 8*hi), col = lane%16
    const int col  = nBase + wave * 16 + mloc;
    const float ba = bias ? bias[col] : 0.0f;
    #pragma unroll
    for (int r = 0; r < 8; ++r) {
        C[(m0 + r + 8 * hi) * OD + col] = acc[r] + ba;
    }
}

// out[b,t,u,o] = lrelu(tnb[b*T+t, o] + pn[b*U+u, o]); bias already folded into tnb.
// grid = (50, T, B), 256 threads, one v4f/thread. Store-bandwidth bound.
__global__ __launch_bounds__(256)
void joint_bcast(const float* __restrict__ tnb,   // [2400][640]
                 const float* __restrict__ pn,    // [640][640]
                 float* __restrict__ out)
{
    const int i  = blockIdx.x * 256 + threadIdx.x;  // 0..12799
    const int u  = i / 160;                          // 0..79
    const int o4 = i - u * 160;                      // 0..159 (v4f index)
    const int bt = blockIdx.z * TD + blockIdx.y;     // 0..2399
    const int pu = blockIdx.z * UD + u;              // 0..639

    const v4f a = *(const v4f*)(tnb + bt * OD + o4 * 4);
    const v4f c = *(const v4f*)(pn  + pu * OD + o4 * 4);
    v4f r = a + c;
    r.x = fmaxf(r.x, 0.01f * r.x);
    r.y = fmaxf(r.y, 0.01f * r.y);
    r.z = fmaxf(r.z, 0.01f * r.z);
    r.w = fmaxf(r.w, 0.01f * r.w);

    v4f* dst = (v4f*)(out + ((size_t)bt * UD + u) * OD) + o4;
    __builtin_nontemporal_store(r, dst);
}

extern "C" void kernel_launch(void* const* d_in, const int* in_sizes, int n_in,
                              void* d_out, int out_size, void* d_ws, size_t ws_size,
                              hipStream_t stream) {
    const float* input_TN = (const float*)d_in[0];   // (8,300,1,512) -> 2400 x 512
    const float* input_PN = (const float*)d_in[1];   // (8,1,80,512)  ->  640 x 512
    const float* W        = (const float*)d_in[2];   // 1024 x 640
    const float* b        = (const float*)d_in[3];   // 640
    float* out = (float*)d_out;

    float* tnb = (float*)d_ws;              // 2400*640 f32 (bias folded)
    float* pnp = tnb + 2400 * OD;           //  640*640 f32

    // tn_proj + bias : 2400x512 @ 512x640
    gemm_wmma_f32<<<dim3(OD / 64, 2400 / 16), dim3(128), 0, stream>>>(
        input_TN, W, b, tnb);
    // pn_proj : 640x512 @ 512x640 (second half of W)
    gemm_wmma_f32<<<dim3(OD / 64, 640 / 16), dim3(128), 0, stream>>>(
        input_PN, W + HD * OD, nullptr, pnp);
    // broadcast add + leaky relu, 491.5 MB NT stores
    joint_bcast<<<dim3(50, TD, BD), dim3(256), 0, stream>>>(tnb, pnp, out);
}